// CustomLayerPCEN2_21036749816206
// MI455X (gfx1250) — compile-verified
//
#include <hip/hip_runtime.h>
#include <math.h>

typedef __attribute__((ext_vector_type(2))) float v2f;
typedef __attribute__((ext_vector_type(4))) float v4f;
typedef __attribute__((ext_vector_type(8))) float v8f;
typedef __attribute__((ext_vector_type(4))) int   v4i;

// typed pointers for the async global->LDS builtin:
// (int4 addrspace(1)* src, int4 addrspace(3)* dst, imm offset, imm cpol)
typedef v4i __attribute__((address_space(1))) gv4i;
typedef v4i __attribute__((address_space(3))) lv4i;

#define PCEN_B 8
#define PCEN_F 128
#define PCEN_T 16384
#define ROW    PCEN_T
#define NCHUNK (PCEN_T / 16)      // 1024 chunks of 16 along T
#define TILE   256                // 16 chunks x 16 steps per WMMA tile
#define NTILE  (PCEN_T / TILE)    // 64 tiles per row
#define THREADS 256
#define NWAVE  (THREADS / 32)     // 8 waves (wave32)

#if defined(__gfx1250__) && __has_builtin(__builtin_amdgcn_wmma_f32_16x16x4_f32)
#define HAVE_WMMA4 1
#else
#define HAVE_WMMA4 0
#endif

#if defined(__gfx1250__) && __has_builtin(__builtin_amdgcn_global_load_async_to_lds_b128)
#define HAVE_ASYNC 1
#else
#define HAVE_ASYNC 0
#endif

__device__ __forceinline__ void wait_async0() {
#if defined(__gfx1250__)
#if __has_builtin(__builtin_amdgcn_s_wait_asynccnt)
  __builtin_amdgcn_s_wait_asynccnt(0);
#else
  asm volatile("s_wait_asynccnt 0x0" ::: "memory");
#endif
#endif
}

// ---------------------------------------------------------------------------
// Kernel 1: per-frequency-row EMA scan (chunked as 16x16 lower-triangular
// WMMA matmuls + LDS carry scan), then write (M+eps)^(-alpha) to workspace.
// One block per row f.  m_t = (1-s) m_{t-1} + s x_t, so within a 16-chunk:
//   m_local = L * x,  L[i,j] = s*(1-s)^(i-j)  (j<=i)
// and the carry from the previous chunk contributes (1-s)^(i+1)*carry.
// ---------------------------------------------------------------------------
__global__ __launch_bounds__(THREADS) void pcen_scan_kernel(
    const float* __restrict__ data,     // [B,F,T]; batch 0 used
    const float* __restrict__ sp,
    const float* __restrict__ alphap,
    const float* __restrict__ epsp,
    float* __restrict__ invm)           // [F,T] = (M+eps)^(-alpha)
{
  __shared__ float sX[ROW];       // row of x0, overwritten in place by m_local
  __shared__ float sFin[NCHUNK];  // per-chunk local finals -> scanned carries

  const int f    = blockIdx.x;
  const int tid  = threadIdx.x;
  const int lane = tid & 31;
  const int wave = tid >> 5;
  const int rlo  = lane & 15;     // row (WMMA) / N-column index
  const int hi   = lane >> 4;     // which half of the wave

  const float s     = sp[0];
  const float alpha = alphap[0];
  const float eps   = epsp[0];
  const float w     = 1.0f - s;

  const float* g = data + (size_t)f * ROW;

  // ---- Phase A: stage the 64KB row into LDS (async global->LDS on gfx1250)
  #pragma unroll
  for (int j = 0; j < ROW / (THREADS * 4); ++j) {
    int e = (tid + j * THREADS) * 4;
#if HAVE_ASYNC
    {
      const void* pg = (const void*)(g + e);
      void*       pl = (void*)(&sX[e]);
      __builtin_amdgcn_global_load_async_to_lds_b128((gv4i*)pg, (lv4i*)pl, 0, 0);
    }
#else
    *(v4f*)(&sX[e]) = *(const v4f*)(g + e);
#endif
  }
#if HAVE_ASYNC
  wait_async0();
#endif
  __syncthreads();

  const float w2 = w * w, w4 = w2 * w2, w8 = w4 * w4, w16 = w8 * w8;

#if HAVE_WMMA4
  // A fragments: L split into four 16x4 K-slices (constant, from s).
  // f32 A layout: lanes 0-15 hold K={4k,4k+1}, lanes 16-31 hold K={4k+2,4k+3}.
  v2f aF[4];
  #pragma unroll
  for (int k = 0; k < 4; ++k) {
    int c0 = 4 * k + 2 * hi;
    float a0 = 0.0f, a1 = 0.0f;
    if (c0 <= rlo)     { a0 = s; for (int p = 0; p < rlo - c0; ++p)     a0 *= w; }
    if (c0 + 1 <= rlo) { a1 = s; for (int p = 0; p < rlo - c0 - 1; ++p) a1 *= w; }
    aF[k].x = a0; aF[k].y = a1;
  }
#endif

  // ---- Phase B: per-tile local scans. Each wave owns its tiles exclusively,
  // so it may overwrite its tile region of sX in place after the matmul.
  for (int tile = wave; tile < NTILE; tile += NWAVE) {
    const int t0 = tile * TILE;
    v8f acc = {};
#if HAVE_WMMA4
    #pragma unroll
    for (int k = 0; k < 4; ++k) {
      // B slice: B[kk,n] = x[t0 + n*16 + kk]; lanes 0-15 K={4k,4k+1}, 16-31 K={4k+2,4k+3}
      int base = t0 + rlo * 16 + 4 * k + 2 * hi;
      v2f b; b.x = sX[base]; b.y = sX[base + 1];
      acc = __builtin_amdgcn_wmma_f32_16x16x4_f32(
          /*neg_a=*/false, aF[k], /*neg_b=*/false, b,
          /*c_mod=*/(short)0, acc, /*reuse_a=*/false, /*reuse_b=*/false);
    }
#else
    // VALU fallback: direct local EMA per output element.
    #pragma unroll
    for (int v = 0; v < 8; ++v) {
      int i = v + 8 * hi;
      float m = 0.0f;
      for (int j = 0; j <= i; ++j) m = w * m + s * sX[t0 + rlo * 16 + j];
      acc[v] = m;
    }
#endif
    // D layout: lane holds D[i,c] for i = v + 8*hi, c = rlo -> 8 contiguous t's.
    int wbase = t0 + rlo * 16 + 8 * hi;
    #pragma unroll
    for (int v = 0; v < 8; ++v) sX[wbase + v] = acc[v];
    if (hi) sFin[tile * 16 + rlo] = acc[7];   // chunk-local final (i = 15)
  }
  __syncthreads();

  // ---- Phase C: inclusive linear-recurrence scan of chunk finals, ratio w^16:
  // S[g] = local_final[g] + w16 * S[g-1]   (Hillis-Steele, 10 steps in LDS)
  float wd = w16;
  for (int d = 1; d < NCHUNK; d <<= 1) {
    float tmp[NCHUNK / THREADS];
    #pragma unroll
    for (int q = 0; q < NCHUNK / THREADS; ++q) {
      int idx = tid + q * THREADS;
      tmp[q] = (idx >= d) ? sFin[idx - d] : 0.0f;
    }
    __syncthreads();
    #pragma unroll
    for (int q = 0; q < NCHUNK / THREADS; ++q) {
      int idx = tid + q * THREADS;
      sFin[idx] += wd * tmp[q];
    }
    __syncthreads();
    wd *= wd;
  }

  // ---- Phase D: apply carries, compute (M+eps)^(-alpha), store coalesced.
  const int i0 = (tid * 4) & 15;          // constant per thread across reps
  float wi = w;                            // w^(i0+1)
  for (int p = 0; p < i0; ++p) wi *= w;
  const float wa = w, wb = w * w, wc = wb * w;
  float* outrow = invm + (size_t)f * ROW;
  #pragma unroll
  for (int rep = 0; rep < ROW / (THREADS * 4); ++rep) {
    int t   = rep * (THREADS * 4) + tid * 4;
    int gch = t >> 4;
    float carry = (gch > 0) ? sFin[gch - 1] : 0.0f;
    float cw = carry * wi;
    float m0 = sX[t + 0] + cw;
    float m1 = sX[t + 1] + cw * wa;
    float m2 = sX[t + 2] + cw * wb;
    float m3 = sX[t + 3] + cw * wc;
    v4f r;
    r.x = powf(m0 + eps, -alpha);
    r.y = powf(m1 + eps, -alpha);
    r.z = powf(m2 + eps, -alpha);
    r.w = powf(m3 + eps, -alpha);
    *(v4f*)(outrow + t) = r;
  }
}

// ---------------------------------------------------------------------------
// Kernel 2: streaming pointwise pass over all B*F*T elements.
// NT hints on the 128MB data/out streams keep the 8MB invm table hot in L2.
// ---------------------------------------------------------------------------
__global__ __launch_bounds__(THREADS) void pcen_pointwise_kernel(
    const float* __restrict__ data,
    const float* __restrict__ rp,
    const float* __restrict__ deltap,
    const float* __restrict__ invm,
    float* __restrict__ out)
{
  const float r_abs = fabsf(rp[0]);
  const float delta = deltap[0];
  const float dpow  = powf(delta, r_abs);

  size_t idx = ((size_t)blockIdx.x * THREADS + threadIdx.x) * 4;
  const size_t rowmask = (size_t)(PCEN_F * PCEN_T) - 1;  // F*T = 2^21

  v4f x  = __builtin_nontemporal_load((const v4f*)(data + idx));
  v4f iv = *(const v4f*)(invm + (idx & rowmask));
  v4f o;
  o.x = powf(x.x * iv.x + delta, r_abs) - dpow;
  o.y = powf(x.y * iv.y + delta, r_abs) - dpow;
  o.z = powf(x.z * iv.z + delta, r_abs) - dpow;
  o.w = powf(x.w * iv.w + delta, r_abs) - dpow;
  __builtin_nontemporal_store(o, (v4f*)(out + idx));
}

// ---------------------------------------------------------------------------
extern "C" void kernel_launch(void* const* d_in, const int* in_sizes, int n_in,
                              void* d_out, int out_size, void* d_ws, size_t ws_size,
                              hipStream_t stream) {
  (void)in_sizes; (void)n_in; (void)out_size; (void)ws_size;
  const float* data  = (const float*)d_in[0];
  const float* alpha = (const float*)d_in[1];
  const float* r     = (const float*)d_in[2];
  const float* delta = (const float*)d_in[3];
  const float* s     = (const float*)d_in[4];
  const float* eps   = (const float*)d_in[5];
  float* out  = (float*)d_out;
  float* invm = (float*)d_ws;   // needs F*T*4 = 8 MB of scratch

  pcen_scan_kernel<<<PCEN_F, THREADS, 0, stream>>>(data, s, alpha, eps, invm);

  const int nblk = (PCEN_B * PCEN_F * PCEN_T) / (THREADS * 4);  // 16384
  pcen_pointwise_kernel<<<nblk, THREADS, 0, stream>>>(data, r, delta, invm, out);
}